// MultiHeadAttention_77833397338310
// MI455X (gfx1250) — compile-verified
//
#include <hip/hip_runtime.h>
#include <hip/hip_bf16.h>

// ---------------------------------------------------------------------------
// MI455X (gfx1250) multi-head-attention forward (qh, attn) producer.
//  * fp32 math via V_WMMA_F32_16X16X4_F32 (wave32 WMMA)
//  * projection GEMM feeds LDS via the Tensor Data Mover (tensor_load_to_lds,
//    TENSORcnt double-buffered pipeline, HW pad -> conflict-free LDS stride)
//  * fused scores+softmax keeps the 16x2048 stripe in 128KB LDS so the
//    268MB attn tensor is written to HBM exactly once.
// ---------------------------------------------------------------------------

typedef float v2f __attribute__((ext_vector_type(2)));
typedef float v8f __attribute__((ext_vector_type(8)));
typedef unsigned int u32x4 __attribute__((ext_vector_type(4)));
typedef int i32x4 __attribute__((ext_vector_type(4)));
typedef int i32x8 __attribute__((ext_vector_type(8)));

__device__ __forceinline__ v8f wmma_f32(v2f a, v2f b, v8f c) {
  // 8 args: (neg_a, A, neg_b, B, c_mod, C, reuse_a, reuse_b)
  return __builtin_amdgcn_wmma_f32_16x16x4_f32(
      false, a, false, b, (short)0, c, false, false);
}

#define B_DIM 2
#define L_DIM 2048
#define NH    8
#define DQK   64
#define DMODEL 512
#define BH    (B_DIM * NH)      // 16
#define BL    (B_DIM * L_DIM)   // 4096
#define SB_STRIDE 2052          // scores stripe row stride (floats)
#define PROJ_LDS_STRIDE 20      // 16 payload + 4 pad DWORDs (TDM pad_amount)
#define PROJ_BUF_FLOATS (64 * PROJ_LDS_STRIDE)
#define NKTILE (DMODEL / 16)    // 32 K-tiles of 16

// --------------------------- TDM support -----------------------------------
#if defined(__has_builtin)
#  if __has_builtin(__builtin_amdgcn_tensor_load_to_lds) && \
      __has_builtin(__builtin_amdgcn_s_wait_tensorcnt)
#    define USE_TDM 1
#  endif
#endif
#ifndef USE_TDM
#  define USE_TDM 0
#endif

#if USE_TDM
// Issue one 2-D TDM tile load: tile_d1 rows x tile_d0 elements (fp32) from a
// row-major tensor with row stride stride_d0, into LDS at lds_byte_off with
// 4 DWORDs of padding inserted after every 16 DWORDs (LDS row stride = 20).
__device__ __forceinline__ void tdm_load_2d(uint32_t lds_byte_off,
                                            const void* gptr,
                                            uint32_t tensor_d0,
                                            uint32_t tensor_d1,
                                            uint32_t stride_d0,
                                            uint32_t tile_d0,
                                            uint32_t tile_d1) {
  const unsigned long long ga = (unsigned long long)(uintptr_t)gptr;
  u32x4 g0;
  g0[0] = 1u;                                   // count=1, user descriptor
  g0[1] = lds_byte_off;                         // lds_addr (bytes)
  g0[2] = (uint32_t)ga;                         // global_addr[31:0]
  g0[3] = ((uint32_t)(ga >> 32) & 0x01FFFFFFu)  // global_addr[56:32]
          | (2u << 30);                         // type = 2 ("image")
  i32x8 g1;
  g1[0] = (int)((2u << 16)      // data_size = 4 bytes
                | (1u << 20)    // pad_enable
                | (3u << 22)    // pad_interval: 16 DWORDs
                | (3u << 25));  // pad_amount:   4 DWORDs
  g1[1] = (int)((tensor_d0 & 0xFFFFu) << 16);                     // dim0 lo16
  g1[2] = (int)((tensor_d0 >> 16) | ((tensor_d1 & 0xFFFFu) << 16));
  g1[3] = (int)((tensor_d1 >> 16) | (tile_d0 << 16));             // tile_dim0
  g1[4] = (int)tile_d1;                                           // tile_dim1
  g1[5] = (int)stride_d0;                                         // stride lo32
  g1[6] = 0;
  g1[7] = 0;
  const i32x4 z4 = {0, 0, 0, 0};
#if __clang_major__ >= 23
  const i32x8 z8 = {0, 0, 0, 0, 0, 0, 0, 0};
  __builtin_amdgcn_tensor_load_to_lds(g0, g1, z4, z4, z8, 0);
#else
  __builtin_amdgcn_tensor_load_to_lds(g0, g1, z4, z4, 0);
#endif
}
#endif  // USE_TDM

// ===========================================================================
// Kernel 1: Y = X @ W^T, scattered into (B, H, L, 64) head layout.
// 64x64 output tile per 256-thread workgroup; 8 waves; wave w does row-block
// (w&3)*16 and two 16-wide col subtiles at (w>>2)*32. A/B K-tiles (64x16 fp32)
// stream into double-buffered LDS via the Tensor Data Mover.
// ===========================================================================
__global__ __launch_bounds__(256) void proj_kernel(
    const float* __restrict__ X,   // (4096, 512) row-major
    const float* __restrict__ W,   // (512, 512) row-major
    float* __restrict__ Out)       // (B, H, L, 64)
{
  __shared__ float As[2][PROJ_BUF_FLOATS];   // 64 rows x 20 (16+pad) floats
  __shared__ float Bs[2][PROJ_BUF_FLOATS];   // W rows kept un-transposed

  const int tid  = threadIdx.x;
  const int wave = tid >> 5;
  const int lane = tid & 31;
  const int l16  = lane & 15;
  const int hi   = lane >> 4;            // 0: K+0/1, 1: K+2/3
  const int rowBlk = (wave & 3) * 16;
  const int colBlk = (wave >> 2) * 32;

  const int tileM = blockIdx.x * 64;
  const int tileN = blockIdx.y * 64;

  v8f acc0 = {};
  v8f acc1 = {};

#if USE_TDM
  const uint32_t asOff = (uint32_t)(uintptr_t)&As[0][0];
  const uint32_t bsOff = (uint32_t)(uintptr_t)&Bs[0][0];
  const uint32_t bufBytes = PROJ_BUF_FLOATS * 4;
  if (wave == 0) {  // prologue: tile 0 -> buffer 0
    tdm_load_2d(asOff, &X[(size_t)tileM * DMODEL], DMODEL, BL, DMODEL, 16, 64);
    tdm_load_2d(bsOff, &W[(size_t)tileN * DMODEL], DMODEL, DMODEL, DMODEL, 16, 64);
  }
#else
  const int ldr = tid >> 2;              // 0..63
  const int ldk = (tid & 3) * 4;         // 0,4,8,12
#endif

  for (int i = 0; i < NKTILE; ++i) {
    const int cur = i & 1;
#if USE_TDM
    if (wave == 0) {
      if (i + 1 < NKTILE) {   // issue next tile into the other buffer
        const int k1 = (i + 1) * 16;
        tdm_load_2d(asOff + (uint32_t)(cur ^ 1) * bufBytes,
                    &X[(size_t)tileM * DMODEL + k1], DMODEL, BL, DMODEL, 16, 64);
        tdm_load_2d(bsOff + (uint32_t)(cur ^ 1) * bufBytes,
                    &W[(size_t)tileN * DMODEL + k1], DMODEL, DMODEL, DMODEL, 16, 64);
        __builtin_amdgcn_s_wait_tensorcnt(2);  // current tile's 2 DMAs done
      } else {
        __builtin_amdgcn_s_wait_tensorcnt(0);
      }
    }
#else
    {  // synchronous fallback fill of buffer `cur`
      const int k0 = i * 16;
      const float4 va = *reinterpret_cast<const float4*>(
          &X[(size_t)(tileM + ldr) * DMODEL + k0 + ldk]);
      float* ar = &As[cur][ldr * PROJ_LDS_STRIDE + ldk];
      ar[0] = va.x; ar[1] = va.y; ar[2] = va.z; ar[3] = va.w;
      const float4 vb = *reinterpret_cast<const float4*>(
          &W[(size_t)(tileN + ldr) * DMODEL + k0 + ldk]);
      float* br = &Bs[cur][ldr * PROJ_LDS_STRIDE + ldk];
      br[0] = vb.x; br[1] = vb.y; br[2] = vb.z; br[3] = vb.w;
    }
#endif
    __syncthreads();

    const float* Ab = &As[cur][0];
    const float* Bb = &Bs[cur][0];
    #pragma unroll
    for (int kk = 0; kk < 16; kk += 4) {
      const int ka = kk + hi * 2;
      const float* arow  = &Ab[(rowBlk + l16) * PROJ_LDS_STRIDE + ka];
      const float* brow0 = &Bb[(colBlk + l16) * PROJ_LDS_STRIDE + ka];
      const float* brow1 = &Bb[(colBlk + 16 + l16) * PROJ_LDS_STRIDE + ka];
      v2f a;  a.x  = arow[0];  a.y  = arow[1];
      v2f b0; b0.x = brow0[0]; b0.y = brow0[1];
      v2f b1; b1.x = brow1[0]; b1.y = brow1[1];
      acc0 = wmma_f32(a, b0, acc0);
      acc1 = wmma_f32(a, b1, acc1);
    }
    __syncthreads();
  }

  // Scatter C: lane<16 -> rows v, lane>=16 -> rows v+8; N = lane&15.
  #pragma unroll
  for (int s = 0; s < 2; ++s) {
    v8f acc = s ? acc1 : acc0;
    const int gcol = tileN + colBlk + s * 16 + l16;
    const int h = gcol >> 6;
    const int d = gcol & 63;
    #pragma unroll
    for (int v = 0; v < 8; ++v) {
      const int grow = tileM + rowBlk + v + hi * 8;
      const int b = grow >> 11;            // /2048
      const int l = grow & 2047;
      Out[(((size_t)(b * NH + h) * L_DIM) + l) * DQK + d] = acc[v];
    }
  }
}

// ===========================================================================
// Kernel 2 (fused): scores -> mask/scale -> row softmax -> attn, one pass.
// Grid: (L/16 row-stripes, B*H). 8 waves each own a 256-col segment.
// The full 16x2048 fp32 score stripe lives in LDS (128KB of the WGP's 320KB),
// so attn is written to HBM exactly once, already normalized.
// ===========================================================================
__global__ __launch_bounds__(256) void scores_softmax_kernel(
    const float* __restrict__ QH,    // (B,H,L,64)
    const float* __restrict__ KH,    // (B,H,L,64)
    const int*   __restrict__ Msk,   // (B,1,L)
    float* __restrict__ Attn)        // (B,H,L,L) softmaxed scores
{
  __shared__ float sb[16 * SB_STRIDE];  // 16 x 2048 score stripe (padded)
  __shared__ float pm[8][16];           // per-wave partial row max
  __shared__ float pl[8][16];           // per-wave partial row sumexp
  __shared__ float rowM[16];
  __shared__ float rowInvL[16];

  const int bh   = blockIdx.y;             // 0..15
  const int b    = bh >> 3;
  const int rb   = blockIdx.x * 16;        // query-row base
  const int tid  = threadIdx.x;
  const int wave = tid >> 5;
  const int lane = tid & 31;
  const int l16  = lane & 15;
  const int hi   = lane >> 4;

  const float* qbase = QH + ((size_t)bh * L_DIM + rb) * DQK;
  const float* kbase = KH + (size_t)bh * L_DIM * DQK;
  float*       abase = Attn + ((size_t)bh * L_DIM + rb) * L_DIM;

  // Preload Q fragments for this 16-row stripe: 16 k-steps of K=4.
  v2f afrag[16];
  {
    const float* arow = qbase + (size_t)l16 * DQK;
    #pragma unroll
    for (int t = 0; t < 16; ++t) {
      const int ka = t * 4 + hi * 2;
      afrag[t].x = arow[ka];
      afrag[t].y = arow[ka + 1];
    }
  }

  float m[8], l[8];
  #pragma unroll
  for (int v = 0; v < 8; ++v) { m[v] = -3.0e38f; l[v] = 0.0f; }

  const int colSeg = wave * 256;
  for (int ct = 0; ct < 16; ++ct) {
    const int c0 = colSeg + ct * 16;
    const float* kcol = kbase + (size_t)(c0 + l16) * DQK;
    if (ct < 15)
      __builtin_prefetch(kbase + (size_t)(c0 + 16 + l16) * DQK, 0, 0);

    v8f acc = {};
    #pragma unroll
    for (int t = 0; t < 16; ++t) {
      const int ka = t * 4 + hi * 2;
      v2f bf; bf.x = kcol[ka]; bf.y = kcol[ka + 1];
      acc = wmma_f32(afrag[t], bf, acc);
    }

    // scale by 1/sqrt(64) and apply key mask (per column = lane&15)
    const int mk = Msk[b * L_DIM + c0 + l16];
    float s[8];
    #pragma unroll
    for (int v = 0; v < 8; ++v) {
      const float x = acc[v] * 0.125f;
      s[v] = (mk == 0) ? -1.0e9f : x;
    }
    // stage raw scores in the LDS stripe
    #pragma unroll
    for (int v = 0; v < 8; ++v) {
      const int grow = v + hi * 8;
      sb[grow * SB_STRIDE + c0 + l16] = s[v];
    }
    // online (max, sumexp) across the 16 lanes holding one row
    #pragma unroll
    for (int v = 0; v < 8; ++v) {
      float tmax = s[v];
      #pragma unroll
      for (int off = 1; off < 16; off <<= 1)
        tmax = fmaxf(tmax, __shfl_xor(tmax, off, 32));
      const float nm = fmaxf(m[v], tmax);
      float p = __expf(s[v] - nm);
      #pragma unroll
      for (int off = 1; off < 16; off <<= 1)
        p += __shfl_xor(p, off, 32);
      l[v] = l[v] * __expf(m[v] - nm) + p;
      m[v] = nm;
    }
  }

  // Publish wave partials.
  if (l16 == 0) {
    #pragma unroll
    for (int v = 0; v < 8; ++v) {
      const int r = v + hi * 8;
      pm[wave][r] = m[v];
      pl[wave][r] = l[v];
    }
  }
  __syncthreads();

  // Finalize per-row (max, 1/sumexp).
  if (tid < 16) {
    float M = -3.0e38f;
    for (int w = 0; w < 8; ++w) M = fmaxf(M, pm[w][tid]);
    float L = 0.0f;
    for (int w = 0; w < 8; ++w) L += pl[w][tid] * __expf(pm[w][tid] - M);
    rowM[tid]    = M;
    rowInvL[tid] = 1.0f / L;
  }
  __syncthreads();

  // Normalized output sweep: 16 rows x 2048 cols, float4 per thread-iter.
  // LDS row starts are 16B aligned (SB_STRIDE*4 % 16 == 0) -> ds_load_b128.
  for (int i = tid; i < 16 * (L_DIM / 4); i += 256) {
    const int r = i >> 9;              // / (2048/4)
    const int c = (i & 511) * 4;
    const float M    = rowM[r];
    const float invL = rowInvL[r];
    float4 x = *reinterpret_cast<const float4*>(&sb[r * SB_STRIDE + c]);
    x.x = __expf(x.x - M) * invL;
    x.y = __expf(x.y - M) * invL;
    x.z = __expf(x.z - M) * invL;
    x.w = __expf(x.w - M) * invL;
    *reinterpret_cast<float4*>(&abase[(size_t)r * L_DIM + c]) = x;
  }
}

// ===========================================================================
extern "C" void kernel_launch(void* const* d_in, const int* in_sizes, int n_in,
                              void* d_out, int out_size, void* d_ws, size_t ws_size,
                              hipStream_t stream) {
  (void)in_sizes; (void)n_in; (void)out_size; (void)ws_size;

  const float* q    = (const float*)d_in[0];
  const float* k    = (const float*)d_in[1];
  // d_in[2] = v   : not needed for (qh, attn)
  const int*   mask = (const int*)d_in[3];
  const float* Wq   = (const float*)d_in[4];
  const float* Wk   = (const float*)d_in[5];
  // d_in[6..9]    : W_v, W_fc, ln_g, ln_b unused by the returned tuple

  float* out  = (float*)d_out;
  float* qh   = out;                                        // B*H*L*64
  float* attn = out + (size_t)BH * L_DIM * DQK;             // B*H*L*L
  float* kh   = (float*)d_ws;                               // B*H*L*64

  dim3 gProj(BL / 64, DMODEL / 64);                         // 64 x 8
  proj_kernel<<<gProj, 256, 0, stream>>>(q, Wq, qh);
  proj_kernel<<<gProj, 256, 0, stream>>>(k, Wk, kh);

  dim3 gScore(L_DIM / 16, BH);                              // 128 x 16
  scores_softmax_kernel<<<gScore, 256, 0, stream>>>(qh, kh, mask, attn);
}